// S4Kernel_74852690035447
// MI455X (gfx1250) — compile-verified
//
#include <hip/hip_runtime.h>
#include <math.h>

#define H_ 1024
#define N_ 64
#define L_ 2048

typedef __attribute__((ext_vector_type(2))) float v2f;
typedef __attribute__((ext_vector_type(8))) float v8f;

__device__ __forceinline__ float fast_rcp(float x) {
  float r = __builtin_amdgcn_rcpf(x);
  return r * (2.0f - x * r);   // one Newton step -> ~full f32 precision
}

// One workgroup per head h. 256 threads = 8 waves (wave32).
__global__ __launch_bounds__(256) void s4_cauchy_woodbury_ifft(
    const float* __restrict__ log_dt,
    const float* __restrict__ Lre, const float* __restrict__ Lim,
    const float* __restrict__ Pre, const float* __restrict__ Pim,
    const float* __restrict__ Bre, const float* __restrict__ Bim,
    const float* __restrict__ Cre, const float* __restrict__ Cim,
    const float* __restrict__ Din,
    float* __restrict__ out)
{
  __shared__ float s_lam_r[N_], s_lam_i[N_];
  __shared__ float s_bmat[2 * N_][16];     // WMMA B matrix (K=128 x N=16), 8KB
  __shared__ float s_kr[L_], s_ki[L_];     // K_hat, then in-place IFFT, 16KB
  __shared__ float s_stage[8][16][18];     // per-wave 16x16 accumulator staging

  const int h   = blockIdx.x;
  const int tid = threadIdx.x;

  // ---------- Phase 1: per-head weights -> LDS ----------
  if (tid < N_) {
    const int n = tid;
    const float xr = Lre[h * N_ + n];
    // lam_re = -softplus(x) computed stably
    const float lr = -(log1pf(expf(-fabsf(xr))) + fmaxf(xr, 0.0f));
    s_lam_r[n] = lr;
    s_lam_i[n] = Lim[h * N_ + n];
    const float pr = Pre[h * N_ + n], pi = Pim[h * N_ + n];
    const float br = Bre[h * N_ + n], bi = Bim[h * N_ + n];
    const float cr = Cre[h * N_ + n], ci = Cim[h * N_ + n];
    // w_j = conj(u) * v for the four einsums
    float w_r[4], w_i[4];
    w_r[0] = pr * br + pi * bi;  w_i[0] = pr * bi - pi * br;  // PRB
    w_r[1] = pr * pr + pi * pi;  w_i[1] = 0.0f;               // PRP
    w_r[2] = cr * br + ci * bi;  w_i[2] = cr * bi - ci * br;  // CRB
    w_r[3] = cr * pr + ci * pi;  w_i[3] = cr * pi - ci * pr;  // CRP
    // B[2n  ][2j] =  wr, B[2n+1][2j] = -wi  -> column 2j   = Re(sum w*R)
    // B[2n  ][2j+1]= wi, B[2n+1][2j+1]= wr  -> column 2j+1 = Im(sum w*R)
#pragma unroll
    for (int j = 0; j < 4; ++j) {
      s_bmat[2 * n    ][2 * j    ] =  w_r[j];
      s_bmat[2 * n + 1][2 * j    ] = -w_i[j];
      s_bmat[2 * n    ][2 * j + 1] =  w_i[j];
      s_bmat[2 * n + 1][2 * j + 1] =  w_r[j];
    }
#pragma unroll
    for (int j = 8; j < 16; ++j) {
      s_bmat[2 * n][j] = 0.0f;
      s_bmat[2 * n + 1][j] = 0.0f;
    }
  }
  __syncthreads();

  const int lane = tid & 31;
  const int wv   = tid >> 5;      // wave id 0..7
  const int half = lane >> 4;     // half-wave selects K-subblock / complex pole
  const int mrow = lane & 15;     // M row within tile == N column for B frag
  const float tdt = 2.0f * expf(-log_dt[h]);   // 2/dt
  const float EPS = 1.1920929e-07f;            // float eps (matches reference guard)

  // Preload tile-invariant fragments into VGPRs:
  //  - bfrag[kk]:  B[K = 2n..2n+1, N = mrow]  (n = 2*kk + half)   -- 64 regs
  //  - lamfrag[kk]: (lam_r[n], lam_i[n]) for this lane's pole     -- 64 regs
  v2f bfrag[32], lamfrag[32];
#pragma unroll
  for (int kk = 0; kk < 32; ++kk) {
    const int n = 2 * kk + half;
    bfrag[kk].x   = s_bmat[2 * n][mrow];
    bfrag[kk].y   = s_bmat[2 * n + 1][mrow];
    lamfrag[kk].x = s_lam_r[n];
    lamfrag[kk].y = s_lam_i[n];
  }

  // ---------- Phase 2: Cauchy reduction via f32 WMMA + Woodbury ----------
  for (int ti = 0; ti < 16; ++ti) {
    const int lbase = (wv * 16 + ti) * 16;
    const int l = lbase + mrow;

    // z = exp(-2*pi*i*l/L); num = 1-z; den = 1+z (guarded); g = (2/dt)*num/den
    const float th = (float)l * (6.283185307179586f / (float)L_);
    float sn, cs;
    sincosf(th, &sn, &cs);
    const float nr = 1.0f - cs, ni = sn;
    float dr = 1.0f + cs, di = -sn;
    float mag2 = dr * dr + di * di;
    if (mag2 < EPS * EPS) { dr = EPS; di = 0.0f; mag2 = EPS * EPS; }
    const float minv = 1.0f / mag2;
    const float gr = tdt * (nr * dr + ni * di) * minv;
    const float gi = tdt * (ni * dr - nr * di) * minv;
    const float facr =  2.0f * dr * minv;      // fac = 2/den
    const float faci = -2.0f * di * minv;

    v8f acc = {0.f, 0.f, 0.f, 0.f, 0.f, 0.f, 0.f, 0.f};
#pragma unroll
    for (int kk = 0; kk < 32; ++kk) {
      const float ddr = gr - lamfrag[kk].x;    // pure-register inner loop
      const float ddi = gi - lamfrag[kk].y;
      const float inv = fast_rcp(ddr * ddr + ddi * ddi);
      v2f a; a.x = ddr * inv; a.y = -ddi * inv;   // A[M=mrow, K=2n..2n+1] = Re,Im of R
      acc = __builtin_amdgcn_wmma_f32_16x16x4_f32(
          false, a, false, bfrag[kk], (short)0, acc, false, false);
    }

    // Stage D tile to per-wave LDS: VGPR r, lane -> (M = r + 8*half, N = mrow)
#pragma unroll
    for (int r = 0; r < 8; ++r)
      s_stage[wv][r + 8 * half][mrow] = acc[r];
    asm volatile("s_wait_dscnt 0" ::: "memory");   // in-wave LDS visibility

    if (half == 0) {     // 16 lanes: one output l each
      const float* st = &s_stage[wv][mrow][0];
      const float prb_r = st[0], prb_i = st[1];
      const float prp_r = st[2], prp_i = st[3];
      const float crb_r = st[4], crb_i = st[5];
      const float crp_r = st[6], crp_i = st[7];
      const float er = 1.0f + prp_r, ei = prp_i;
      const float einv = 1.0f / (er * er + ei * ei);
      const float qr = (prb_r * er + prb_i * ei) * einv;   // PRB/(1+PRP)
      const float qi = (prb_i * er - prb_r * ei) * einv;
      const float kr = crb_r - (crp_r * qr - crp_i * qi);  // CRB - CRP*q
      const float ki = crb_i - (crp_r * qi + crp_i * qr);
      s_kr[l] = facr * kr - faci * ki;                     // K_hat = fac*(...)
      s_ki[l] = facr * ki + faci * kr;
    }
  }
  __syncthreads();

  // ---------- Phase 3: 2048-pt inverse FFT in LDS (DIF, nat -> bitrev) ----------
#pragma unroll 1
  for (int s = 11; s >= 1; --s) {
    const int hlf = 1 << (s - 1);
    for (int q = tid; q < (L_ / 2); q += 256) {
      const int j  = q & (hlf - 1);
      const int i0 = ((q >> (s - 1)) << s) + j;
      const int i1 = i0 + hlf;
      const float ar = s_kr[i0], ai = s_ki[i0];
      const float br2 = s_kr[i1], bi2 = s_ki[i1];
      const float tr = ar - br2, ti2 = ai - bi2;
      s_kr[i0] = ar + br2;
      s_ki[i0] = ai + bi2;
      const float ang = (float)j * (6.283185307179586f / (float)(1 << s));
      float ws, wc;
      sincosf(ang, &ws, &wc);                 // inverse: +angle twiddle
      s_kr[i1] = tr * wc - ti2 * ws;
      s_ki[i1] = tr * ws + ti2 * wc;
    }
    __syncthreads();
  }

  // Coalesced stores; undo bit-reversal on the LDS read side. Output = Re(ifft) (1/L scale).
  for (int p = tid; p < L_; p += 256) {
    const int src = (int)(__brev((unsigned)p) >> 21);    // 11-bit reverse
    out[h * L_ + p] = s_kr[src] * (1.0f / (float)L_);
  }
  if (tid == 0) out[H_ * L_ + h] = Din[h];   // second output: D pass-through
}

extern "C" void kernel_launch(void* const* d_in, const int* in_sizes, int n_in,
                              void* d_out, int out_size, void* d_ws, size_t ws_size,
                              hipStream_t stream) {
  (void)in_sizes; (void)n_in; (void)out_size; (void)d_ws; (void)ws_size;
  const float* log_dt = (const float*)d_in[0];
  const float* Lre    = (const float*)d_in[1];
  const float* Lim    = (const float*)d_in[2];
  const float* Pre    = (const float*)d_in[3];
  const float* Pim    = (const float*)d_in[4];
  const float* Bre    = (const float*)d_in[5];
  const float* Bim    = (const float*)d_in[6];
  const float* Cre    = (const float*)d_in[7];
  const float* Cim    = (const float*)d_in[8];
  const float* Din    = (const float*)d_in[9];
  float* out = (float*)d_out;
  s4_cauchy_woodbury_ifft<<<H_, 256, 0, stream>>>(
      log_dt, Lre, Lim, Pre, Pim, Bre, Bim, Cre, Cim, Din, out);
}